// HybridModel_42631845380853
// MI455X (gfx1250) — compile-verified
//
#include <hip/hip_runtime.h>

typedef __attribute__((ext_vector_type(2))) float v2f;
typedef __attribute__((ext_vector_type(8))) float v8f;

__device__ __forceinline__ v8f wmma_f32_16x16x4(v2f a, v2f b, v8f c) {
  // 8 args: (neg_a, A, neg_b, B, c_mod, C, reuse_a, reuse_b)
  return __builtin_amdgcn_wmma_f32_16x16x4_f32(false, a, false, b, (short)0, c,
                                               false, false);
}

// ds_swizzle group-of-32: offset = {1'b0, xor[4:0], or[4:0], and[4:0]}
__device__ __forceinline__ float swz_xor8(float v) {
  return __int_as_float(
      __builtin_amdgcn_ds_swizzle(__float_as_int(v), 0x201f));  // SWAPX8
}

// ---------------- build the fixed 16x16 circuit matrix ----------------
// Thread t simulates basis column |t>, applying the shared trainable gates.
// Wire w acts on bit (3-w) (MSB-first), so wire w -> mask 8>>w.

template <int MASK>
__device__ __forceinline__ void apply1q(float* sr, float* si,
    float m00r, float m00i, float m01r, float m01i,
    float m10r, float m10i, float m11r, float m11i) {
#pragma unroll
  for (int k = 0; k < 16; ++k) {
    if ((k & MASK) == 0) {
      const int k1 = k | MASK;
      float ar = sr[k], ai = si[k], br = sr[k1], bi = si[k1];
      sr[k]  = m00r * ar - m00i * ai + m01r * br - m01i * bi;
      si[k]  = m00r * ai + m00i * ar + m01r * bi + m01i * br;
      sr[k1] = m10r * ar - m10i * ai + m11r * br - m11i * bi;
      si[k1] = m10r * ai + m10i * ar + m11r * bi + m11i * br;
    }
  }
}

template <int CM, int TM>
__device__ __forceinline__ void applyCtrl(float* sr, float* si,
    float m00r, float m00i, float m01r, float m01i,
    float m10r, float m10i, float m11r, float m11i) {
#pragma unroll
  for (int k = 0; k < 16; ++k) {
    if ((k & CM) != 0 && (k & TM) == 0) {
      const int k1 = k | TM;
      float ar = sr[k], ai = si[k], br = sr[k1], bi = si[k1];
      sr[k]  = m00r * ar - m00i * ai + m01r * br - m01i * bi;
      si[k]  = m00r * ai + m00i * ar + m01r * bi + m01i * br;
      sr[k1] = m10r * ar - m10i * ai + m11r * br - m11i * bi;
      si[k1] = m10r * ai + m10i * ar + m11r * bi + m11i * br;
    }
  }
}

__global__ void build_u_kernel(const float* __restrict__ rxp,
                               const float* __restrict__ ryp,
                               const float* __restrict__ rzp,
                               const float* __restrict__ crxp,
                               float* __restrict__ wsBr,
                               float* __restrict__ wsBi) {
  const int t = threadIdx.x;
  if (t >= 16) return;
  float sr[16], si[16];
#pragma unroll
  for (int j = 0; j < 16; ++j) { sr[j] = (j == t) ? 1.0f : 0.0f; si[j] = 0.0f; }

#define GATES_FOR_WIRE(W, MASK)                                                  \
  { float h = rxp[W] * 0.5f, c = cosf(h), s = sinf(h);                           \
    apply1q<MASK>(sr, si, c, 0.f, 0.f, -s, 0.f, -s, c, 0.f); }                   \
  { float h = ryp[W] * 0.5f, c = cosf(h), s = sinf(h);                           \
    apply1q<MASK>(sr, si, c, 0.f, -s, 0.f, s, 0.f, c, 0.f); }                    \
  { float h = rzp[W] * 0.5f, c = cosf(h), s = sinf(h);                           \
    apply1q<MASK>(sr, si, c, -s, 0.f, 0.f, 0.f, 0.f, c, s); }

  GATES_FOR_WIRE(0, 8)
  GATES_FOR_WIRE(1, 4)
  GATES_FOR_WIRE(2, 2)
  GATES_FOR_WIRE(3, 1)
#undef GATES_FOR_WIRE

#define ENT_FOR(I, CM, TM)                                                       \
  { float h = crxp[I] * 0.5f, c = cosf(h), s = sinf(h);                          \
    applyCtrl<CM, TM>(sr, si, c, 0.f, 0.f, -s, 0.f, -s, c, 0.f); }               \
  applyCtrl<CM, TM>(sr, si, 0.f, 0.f, 1.f, 0.f, 1.f, 0.f, 0.f, 0.f);

  ENT_FOR(0, 8, 4)
  ENT_FOR(1, 4, 2)
  ENT_FOR(2, 2, 1)
#undef ENT_FOR

  // Fold encoding diagonal: U'[:,t] = (-i)^popcount(t) * U[:,t]
  const int p = __popc(t) & 3;
  // B fragment layout for V_WMMA_F32_16X16X4_F32 (K-chunks of 4):
  //   column K=t lives in chunk c=t>>2; within-chunk kk=t&3 -> vgpr v=kk&1,
  //   lane-half hi=kk>>1; element B[K][N] at ws[(2c+v)*32 + hi*16 + N].
  const int c  = t >> 2;
  const int dl = t & 3;
  const int vg = 2 * c + (dl & 1);
  const int hb = (dl >> 1) * 16;
#pragma unroll
  for (int j = 0; j < 16; ++j) {
    float ur = sr[j], ui = si[j], wr, wi;
    if (p == 0)      { wr = ur;  wi = ui;  }
    else if (p == 1) { wr = ui;  wi = -ur; }   // * (-i)
    else if (p == 2) { wr = -ur; wi = -ui; }   // * (-1)
    else             { wr = -ui; wi = ur;  }   // * (i)
    wsBr[vg * 32 + hb + j] = wr;
    wsBi[vg * 32 + hb + j] = wi;
  }
}

// ---------------- main batched kernel ----------------
// One wave handles a tile of 16 samples:
//   D[sample][amp] = A[sample][k] x B[k][amp]   (two real f32 WMMA chains)

#define TROW 20  // prob-transpose row stride in dwords (bank-friendly, 16B rows)

__global__ __launch_bounds__(256) void qnn_kernel(
    const float* __restrict__ x,
    const float* __restrict__ fc1_w, const float* __restrict__ fc1_b,
    const float* __restrict__ fc2_w, const float* __restrict__ fc2_b,
    const float* __restrict__ wsBr, const float* __restrict__ wsBi,
    float* __restrict__ out, int nSamples, int iters) {
  __shared__ float lBr[256];
  __shared__ float lBi[256];
  __shared__ float4 lW1[64];                       // fc1 rows
  __shared__ float4 lWb[64];                       // (b1, w2r0, w2r1, 0)
  __shared__ __align__(16) float lT[8 * 16 * TROW];  // prob transpose scratch

  const int tid = threadIdx.x;
  lBr[tid] = wsBr[tid];
  lBi[tid] = wsBi[tid];
  if (tid < 64) {
    lW1[tid] = make_float4(fc1_w[tid * 4 + 0], fc1_w[tid * 4 + 1],
                           fc1_w[tid * 4 + 2], fc1_w[tid * 4 + 3]);
    lWb[tid] = make_float4(fc1_b[tid], fc2_w[tid], fc2_w[64 + tid], 0.0f);
  }
  __syncthreads();

  const int lane = tid & 31;
  const int wave = tid >> 5;
  const int hi   = lane >> 4;
  const int jn   = lane & 15;  // amplitude index in C/D layout

  // Hoist B fragments into VGPRs for the whole loop.
  v2f BR[4], BI[4];
#pragma unroll
  for (int c = 0; c < 4; ++c) {
    BR[c][0] = lBr[(2 * c + 0) * 32 + lane];
    BR[c][1] = lBr[(2 * c + 1) * 32 + lane];
    BI[c][0] = lBi[(2 * c + 0) * 32 + lane];
    BI[c][1] = lBi[(2 * c + 1) * 32 + lane];
  }

  const float b20 = fc2_b[0];
  const float b21 = fc2_b[1];

  const int oh = jn >> 3;            // which half of the 64 fc1 outputs
  const int sl = hi * 8 + (jn & 7);  // sample-in-tile for the MLP stage
  float* const tStore = &lT[wave * 16 * TROW + hi * 8 * TROW + jn];
  const float4* const tLoad =
      (const float4*)&lT[wave * 16 * TROW + sl * TROW];

  for (int it = 0; it < iters; ++it) {
    const int tile = (blockIdx.x * 8 + wave) * iters + it;
    const int base = tile * 16;
    if (base + 16 > nSamples) break;  // uniform per wave

    // ---- encoding amplitudes m[k] (A fragment) ----
    const float4 xs = ((const float4*)x)[base + jn];
    const float c0 = cosf(xs.x * 0.5f), s0v = sinf(xs.x * 0.5f);
    const float c1 = cosf(xs.y * 0.5f), s1v = sinf(xs.y * 0.5f);
    const float c2 = cosf(xs.z * 0.5f), s2v = sinf(xs.z * 0.5f);
    const float c3 = cosf(xs.w * 0.5f), s3v = sinf(xs.w * 0.5f);
    const float f2 = hi ? s2v : c2;  // bit1 of k == lane-half

    v2f A[4];
#pragma unroll
    for (int c = 0; c < 4; ++c) {
      const float f0   = (c & 2) ? s0v : c0;  // bit3 of k
      const float f1   = (c & 1) ? s1v : c1;  // bit2 of k
      const float f012 = f0 * f1 * f2;
      A[c][0] = f012 * c3;                    // bit0 = 0
      A[c][1] = f012 * s3v;                   // bit0 = 1
    }

    // ---- two real f32 WMMA chains over K=16 ----
    v8f accR = {0, 0, 0, 0, 0, 0, 0, 0};
    v8f accI = {0, 0, 0, 0, 0, 0, 0, 0};
#pragma unroll
    for (int c = 0; c < 4; ++c) {
      accR = wmma_f32_16x16x4(A[c], BR[c], accR);
      accI = wmma_f32_16x16x4(A[c], BI[c], accI);
    }

    // ---- probs -> LDS transpose (lane j, vgpr r  ->  row sample, col j) ----
#pragma unroll
    for (int r = 0; r < 8; ++r) {
      const float pr = accR[r] * accR[r] + accI[r] * accI[r];
      tStore[r * TROW] = pr;
    }
    __builtin_amdgcn_wave_barrier();  // same-wave DS ops are in-order

    const float4 fA = tLoad[0];
    const float4 fB = tLoad[1];
    const float4 fC = tLoad[2];
    const float4 fD = tLoad[3];
    __builtin_amdgcn_wave_barrier();

    // ---- PauliZ expectations via shared Walsh adder tree ----
    // q3 (bit0 signs): sum of within-pair differences
    const float q3 = ((fA.x - fA.y) + (fA.z - fA.w)) +
                     ((fB.x - fB.y) + (fB.z - fB.w)) +
                     ((fC.x - fC.y) + (fC.z - fC.w)) +
                     ((fD.x - fD.y) + (fD.z - fD.w));
    const float s0 = fA.x + fA.y, s1 = fA.z + fA.w;
    const float s2 = fB.x + fB.y, s3 = fB.z + fB.w;
    const float s4 = fC.x + fC.y, s5 = fC.z + fC.w;
    const float s6 = fD.x + fD.y, s7 = fD.z + fD.w;
    const float q2 = (s0 - s1) + (s2 - s3) + (s4 - s5) + (s6 - s7);  // bit1
    const float t0 = s0 + s1, t1 = s2 + s3, t2 = s4 + s5, t3 = s6 + s7;
    const float q1 = (t0 - t1) + (t2 - t3);                          // bit2
    const float q0 = (t0 + t1) - (t2 + t3);                          // bit3

    // ---- fc1 (ReLU) + fc2 partials (each lane: 32 hidden units) ----
    float pl0 = 0.f, pl1 = 0.f;
#pragma unroll
    for (int ol = 0; ol < 32; ++ol) {
      const int o = oh * 32 + ol;
      const float4 wr = lW1[o];
      const float4 wb = lWb[o];
      float h = fmaf(q0, wr.x,
                fmaf(q1, wr.y, fmaf(q2, wr.z, fmaf(q3, wr.w, wb.x))));
      h = fmaxf(h, 0.f);
      pl0 = fmaf(h, wb.y, pl0);
      pl1 = fmaf(h, wb.z, pl1);
    }
    pl0 += swz_xor8(pl0);  // combine the two output halves (lanes l <-> l^8)
    pl1 += swz_xor8(pl1);
    pl0 += b20;
    pl1 += b21;

    const float mx  = fmaxf(pl0, pl1);
    const float lse = mx + logf(expf(pl0 - mx) + expf(pl1 - mx));
    if ((lane & 8) == 0) {
      ((float2*)out)[base + sl] = make_float2(pl0 - lse, pl1 - lse);
    }
  }
}

extern "C" void kernel_launch(void* const* d_in, const int* in_sizes, int n_in,
                              void* d_out, int out_size, void* d_ws,
                              size_t ws_size, hipStream_t stream) {
  (void)n_in; (void)out_size; (void)ws_size;
  const float* x     = (const float*)d_in[0];
  const float* rxp   = (const float*)d_in[1];
  const float* ryp   = (const float*)d_in[2];
  const float* rzp   = (const float*)d_in[3];
  const float* crxp  = (const float*)d_in[4];
  const float* fc1_w = (const float*)d_in[5];
  const float* fc1_b = (const float*)d_in[6];
  const float* fc2_w = (const float*)d_in[7];
  const float* fc2_b = (const float*)d_in[8];
  float* out = (float*)d_out;

  float* wsBr = (float*)d_ws;       // 256 floats
  float* wsBi = wsBr + 256;         // 256 floats

  build_u_kernel<<<1, 32, 0, stream>>>(rxp, ryp, rzp, crxp, wsBr, wsBi);

  const int nSamples = in_sizes[0] / 4;
  const int iters = 4;                       // samples per block = 128*iters
  const int spb = 128 * iters;
  const int blocks = (nSamples + spb - 1) / spb;
  qnn_kernel<<<blocks, 256, 0, stream>>>(x, fc1_w, fc1_b, fc2_w, fc2_b,
                                         wsBr, wsBi, out, nSamples, iters);
}